// SeqEncoding_45973329937156
// MI455X (gfx1250) — compile-verified
//
#include <hip/hip_runtime.h>

// Problem constants (match reference)
#define BB     4
#define NN     512
#define HH     256
#define NHEADS 8
#define SUBH   32
#define FFD    1024
#define TOK    (BB * NN)   // 2048 tokens

typedef float v2f __attribute__((ext_vector_type(2)));
typedef float v8f __attribute__((ext_vector_type(8)));

// ---------------------------------------------------------------------------
// Pair-packed B layout: P[(k>>1)*Nc + n] = float2( B[k][n], B[k+1][n] ).
// The f32 WMMA B fragment needs rows {k+2hf, k+2hf+1} at one column -> one
// b64 load from the packed layout instead of two strided b32 loads.
//
// One wave computes a 16x16 fp32 tile D = A[16xK] * B[Kx16] with
// V_WMMA_F32_16X16X4_F32 (exact fp32 accumulate). A is natural row-major
// (lda, even element count), B is pair-packed with Nc float2 per K-pair row;
// Bp must already include the column-tile offset (in float2 units).
// NTA: mark A loads non-temporal (for stream-once data like T, so the 192MB
// L2 keeps reusable data — attn, packed weights — resident instead).
// ---------------------------------------------------------------------------
template <int K, bool NTA>
__device__ __forceinline__ v8f wmma_tile_f32p(const float* __restrict__ A, int lda,
                                              const v2f* __restrict__ Bp, int nc) {
  const int lane = threadIdx.x & 31;
  const int hf   = lane >> 4;     // half-wave selector
  const int l15  = lane & 15;
  v8f c = {0.f, 0.f, 0.f, 0.f, 0.f, 0.f, 0.f, 0.f};
  const v2f* ap = (const v2f*)(A + (size_t)l15 * lda + 2 * hf);  // 8B aligned
  const v2f* bp = Bp + (size_t)hf * nc + l15;
#pragma unroll 8
  for (int k = 0; k < K; k += 4) {
    v2f a, b;
    if (NTA)
      a = __builtin_nontemporal_load(ap + (k >> 1));
    else
      a = ap[k >> 1];
    b = bp[(size_t)(k >> 1) * nc];
    c = __builtin_amdgcn_wmma_f32_16x16x4_f32(false, a, false, b, (short)0, c,
                                              false, false);
  }
  return c;
}

// ---------------------------------------------------------------------------
// 0a) pad Wqk [256 x 8] -> pair-packed [128][16] float2 with zero pad cols
// ---------------------------------------------------------------------------
__global__ __launch_bounds__(256) void k_pack_wqk(const float* __restrict__ Wqk,
                                                  float* __restrict__ wqkp) {
  const int idx = blockIdx.x * blockDim.x + threadIdx.x;   // over 256*16
  const int k = idx >> 4, n = idx & 15;
  const float v = (n < NHEADS) ? Wqk[k * NHEADS + n] : 0.f;
  wqkp[((size_t)(k >> 1) * 16 + n) * 2 + (k & 1)] = v;
}

// 0b) generic pair-pack: src [K x Nc] row-major -> dst [(K/2) x Nc] float2
__global__ __launch_bounds__(256) void k_pack_pairs(const float* __restrict__ src,
                                                    float* __restrict__ dst,
                                                    int K, int Nc) {
  const int idx = blockIdx.x * blockDim.x + threadIdx.x;
  if (idx >= K * Nc) return;
  const int k = idx / Nc, n = idx % Nc;
  dst[((size_t)(k >> 1) * Nc + n) * 2 + (k & 1)] = src[idx];
}

// ---------------------------------------------------------------------------
// 1) attn logits: attn[b,h,i,j] = sum_e T[b,i,j,e]*Wqk[e,h] + maskadd
//    One wave per (b, i, 16-wide j tile); B is padded/pair-packed Wqk, so
//    the inner loop is branch-free: NT b64(A from T) + b64(B) + wmma.
// ---------------------------------------------------------------------------
__global__ __launch_bounds__(32) void k_scores(const float* __restrict__ T,
                                               const float* __restrict__ wqkp,
                                               const unsigned char* __restrict__ masks,
                                               float* __restrict__ attn) {
  const int jt = blockIdx.x;           // j tile (0..31)
  const int i  = blockIdx.y;           // 0..511
  const int b  = blockIdx.z;           // 0..3
  const int lane = threadIdx.x & 31;
  const int hf = lane >> 4, l15 = lane & 15;

  const float* A = T + (((size_t)b * NN + i) * NN + (size_t)jt * 16) * HH;
  v8f c = wmma_tile_f32p<HH, true>(A, HH, (const v2f*)wqkp, 16);

  if (l15 < NHEADS) {
    const float mi = masks[b * NN + i] ? 1.f : 0.f;
    float* orow = attn + (((size_t)b * NHEADS + l15) * NN + i) * NN;
#pragma unroll
    for (int r = 0; r < 8; ++r) {
      const int j = jt * 16 + hf * 8 + r;   // D tile: M = r (+8 upper half)
      const float mj = masks[b * NN + j] ? 1.f : 0.f;
      orow[j] = c[r] - 1000.f * (1.f - mi * mj);
    }
  }
}

// ---------------------------------------------------------------------------
// 2) softmax over last axis (512), one wave per row, 16 elems/lane (wave32).
// ---------------------------------------------------------------------------
__global__ __launch_bounds__(256) void k_softmax(float* __restrict__ attn) {
  const int row  = blockIdx.x * (blockDim.x >> 5) + (threadIdx.x >> 5);
  const int lane = threadIdx.x & 31;
  float* p = attn + (size_t)row * NN;

  float v[16];
  float m = -3.4e38f;
#pragma unroll
  for (int t = 0; t < 16; ++t) { v[t] = p[lane + t * 32]; m = fmaxf(m, v[t]); }
#pragma unroll
  for (int off = 16; off; off >>= 1) m = fmaxf(m, __shfl_xor(m, off, 32));
  float s = 0.f;
#pragma unroll
  for (int t = 0; t < 16; ++t) { v[t] = __expf(v[t] - m); s += v[t]; }
#pragma unroll
  for (int off = 16; off; off >>= 1) s += __shfl_xor(s, off, 32);
  const float inv = 1.f / s;
#pragma unroll
  for (int t = 0; t < 16; ++t) p[lane + t * 32] = v[t] * inv;
}

// ---------------------------------------------------------------------------
// 3) v = S @ Wv + bv, emitted PAIR-PACKED per (b,h): for each head,
//    vws[bh][(j>>1)*SUBH + d] is float2(v[j], v[j+1]) so it is directly the
//    B operand of the ctx GEMM.
// ---------------------------------------------------------------------------
__global__ __launch_bounds__(32) void k_vproj(const float* __restrict__ S,
                                              const float* __restrict__ wvp,
                                              const float* __restrict__ bv,
                                              float* __restrict__ vws) {
  const int mt = blockIdx.x;   // token tile (0..127)
  const int nt = blockIdx.y;   // H tile (0..15)
  const int lane = threadIdx.x & 31;
  const int hf = lane >> 4, l15 = lane & 15;

  v8f c = wmma_tile_f32p<HH, false>(S + (size_t)mt * 16 * HH, HH,
                                    (const v2f*)wvp + nt * 16, HH);

  const int n = nt * 16 + l15;
  const int h = n / SUBH, d = n % SUBH;
  const float bias = bv[n];
#pragma unroll
  for (int r = 0; r < 8; ++r) {
    const int row = mt * 16 + hf * 8 + r;    // global token
    const int b = row / NN, j = row % NN;
    const size_t bh = (size_t)b * NHEADS + h;
    vws[bh * NN * SUBH + (((size_t)(j >> 1) * SUBH + d) * 2 + (j & 1))] =
        c[r] + bias;
  }
}

// ---------------------------------------------------------------------------
// 4) ctx[b,i,h*32+d] = sum_j attn[b,h,i,j] * v[b,h,j,d]   (K = 512)
// ---------------------------------------------------------------------------
__global__ __launch_bounds__(32) void k_ctx(const float* __restrict__ attn,
                                            const float* __restrict__ vws,
                                            float* __restrict__ ctxws) {
  const int it = blockIdx.x;   // i tile (0..31)
  const int dt = blockIdx.y;   // d tile (0..1)
  const int bh = blockIdx.z;   // 0..31
  const int b = bh >> 3, h = bh & 7;
  const int lane = threadIdx.x & 31;
  const int hf = lane >> 4, l15 = lane & 15;

  const float* A = attn + ((size_t)bh * NN + (size_t)it * 16) * NN;   // lda = NN
  const v2f* Bp = (const v2f*)vws + (size_t)bh * (NN / 2) * SUBH + dt * 16;
  v8f c = wmma_tile_f32p<NN, false>(A, NN, Bp, SUBH);

  const int d = dt * 16 + l15;
#pragma unroll
  for (int r = 0; r < 8; ++r) {
    const int i = it * 16 + hf * 8 + r;
    ctxws[((size_t)b * NN + i) * HH + h * SUBH + d] = c[r];
  }
}

// ---------------------------------------------------------------------------
// 5) pre0 = S + relu(ctx @ Wo + bo)
// ---------------------------------------------------------------------------
__global__ __launch_bounds__(32) void k_attnout(const float* __restrict__ ctxws,
                                                const float* __restrict__ wop,
                                                const float* __restrict__ bo,
                                                const float* __restrict__ S,
                                                float* __restrict__ pre0) {
  const int mt = blockIdx.x, nt = blockIdx.y;
  const int lane = threadIdx.x & 31;
  const int hf = lane >> 4, l15 = lane & 15;

  v8f c = wmma_tile_f32p<HH, false>(ctxws + (size_t)mt * 16 * HH, HH,
                                    (const v2f*)wop + nt * 16, HH);

  const int n = nt * 16 + l15;
  const float bias = bo[n];
#pragma unroll
  for (int r = 0; r < 8; ++r) {
    const int row = mt * 16 + hf * 8 + r;
    float t = fmaxf(c[r] + bias, 0.f);
    pre0[(size_t)row * HH + n] = t + S[(size_t)row * HH + n];
  }
}

// ---------------------------------------------------------------------------
// 6/9) LayerNorm over 256 cols, one wave per row (8 elems/lane).
// ---------------------------------------------------------------------------
__global__ __launch_bounds__(256) void k_layernorm(const float* __restrict__ x,
                                                   const float* __restrict__ g,
                                                   const float* __restrict__ beta,
                                                   float* __restrict__ y) {
  const int row  = blockIdx.x * (blockDim.x >> 5) + (threadIdx.x >> 5);
  const int lane = threadIdx.x & 31;
  const float* p = x + (size_t)row * HH;

  float v[8];
  float s = 0.f;
#pragma unroll
  for (int t = 0; t < 8; ++t) { v[t] = p[lane + t * 32]; s += v[t]; }
#pragma unroll
  for (int off = 16; off; off >>= 1) s += __shfl_xor(s, off, 32);
  const float mean = s * (1.f / HH);
  float q = 0.f;
#pragma unroll
  for (int t = 0; t < 8; ++t) { const float d = v[t] - mean; q += d * d; }
#pragma unroll
  for (int off = 16; off; off >>= 1) q += __shfl_xor(q, off, 32);
  const float rstd = rsqrtf(q * (1.f / HH) + 1e-5f);
#pragma unroll
  for (int t = 0; t < 8; ++t) {
    const int col = lane + t * 32;
    y[(size_t)row * HH + col] = (v[t] - mean) * rstd * g[col] + beta[col];
  }
}

// ---------------------------------------------------------------------------
// 7) h1 = relu(x @ W1 + b1)   [2048 x 1024], K = 256
// ---------------------------------------------------------------------------
__global__ __launch_bounds__(32) void k_ffn1(const float* __restrict__ xws,
                                             const float* __restrict__ w1p,
                                             const float* __restrict__ b1,
                                             float* __restrict__ h1ws) {
  const int mt = blockIdx.x, nt = blockIdx.y;
  const int lane = threadIdx.x & 31;
  const int hf = lane >> 4, l15 = lane & 15;

  v8f c = wmma_tile_f32p<HH, false>(xws + (size_t)mt * 16 * HH, HH,
                                    (const v2f*)w1p + nt * 16, FFD);

  const int n = nt * 16 + l15;
  const float bias = b1[n];
#pragma unroll
  for (int r = 0; r < 8; ++r) {
    const int row = mt * 16 + hf * 8 + r;
    h1ws[(size_t)row * FFD + n] = fmaxf(c[r] + bias, 0.f);
  }
}

// ---------------------------------------------------------------------------
// 8) pre1 = x + h1 @ W2 + b2   [2048 x 256], K = 1024
// ---------------------------------------------------------------------------
__global__ __launch_bounds__(32) void k_ffn2(const float* __restrict__ h1ws,
                                             const float* __restrict__ w2p,
                                             const float* __restrict__ b2,
                                             const float* __restrict__ xws,
                                             float* __restrict__ pre1) {
  const int mt = blockIdx.x, nt = blockIdx.y;
  const int lane = threadIdx.x & 31;
  const int hf = lane >> 4, l15 = lane & 15;

  v8f c = wmma_tile_f32p<FFD, false>(h1ws + (size_t)mt * 16 * FFD, FFD,
                                     (const v2f*)w2p + nt * 16, HH);

  const int n = nt * 16 + l15;
  const float bias = b2[n];
#pragma unroll
  for (int r = 0; r < 8; ++r) {
    const int row = mt * 16 + hf * 8 + r;
    pre1[(size_t)row * HH + n] = c[r] + bias + xws[(size_t)row * HH + n];
  }
}

// ---------------------------------------------------------------------------
// Launcher
// ---------------------------------------------------------------------------
extern "C" void kernel_launch(void* const* d_in, const int* in_sizes, int n_in,
                              void* d_out, int out_size, void* d_ws, size_t ws_size,
                              hipStream_t stream) {
  (void)in_sizes; (void)n_in; (void)out_size; (void)ws_size;

  const float* S   = (const float*)d_in[0];
  const float* T   = (const float*)d_in[1];
  const unsigned char* masks = (const unsigned char*)d_in[2];
  const float* Wqk = (const float*)d_in[3];
  const float* Wv  = (const float*)d_in[4];
  const float* bv  = (const float*)d_in[5];
  const float* Wo  = (const float*)d_in[6];
  const float* bo  = (const float*)d_in[7];
  const float* g0  = (const float*)d_in[8];
  const float* be0 = (const float*)d_in[9];
  const float* W1  = (const float*)d_in[10];
  const float* b1  = (const float*)d_in[11];
  const float* W2  = (const float*)d_in[12];
  const float* b2  = (const float*)d_in[13];
  const float* g1  = (const float*)d_in[14];
  const float* be1 = (const float*)d_in[15];

  float* out  = (float*)d_out;                       // [B,N,H]
  float* attn = out + (size_t)BB * NN * HH;          // [B,heads,N,N]

  float* ws = (float*)d_ws;
  float* wqkp  = ws;                                 // 256*16        = 4096
  float* wvp   = wqkp  + (size_t)4096;               // 256*256       = 65536
  float* wop   = wvp   + (size_t)65536;              // 256*256       = 65536
  float* w1p   = wop   + (size_t)65536;              // 256*1024      = 262144
  float* w2p   = w1p   + (size_t)262144;             // 1024*256      = 262144
  float* vws   = w2p   + (size_t)262144;             // B*H*N*SUBH    = 524288
  float* ctxws = vws   + (size_t)524288;             // TOK*HH        = 524288
  float* pre0  = ctxws + (size_t)524288;             // TOK*HH        = 524288
  float* xws   = pre0  + (size_t)524288;             // TOK*HH        = 524288
  float* h1ws  = xws   + (size_t)524288;             // TOK*FFD       = 2097152
  float* pre1  = h1ws  + (size_t)2097152;            // TOK*HH        = 524288

  // 0) pair-pack weights (one-time per call, tiny)
  k_pack_wqk<<<dim3((256 * 16) / 256), 256, 0, stream>>>(Wqk, wqkp);
  k_pack_pairs<<<dim3((HH * HH) / 256), 256, 0, stream>>>(Wv, wvp, HH, HH);
  k_pack_pairs<<<dim3((HH * HH) / 256), 256, 0, stream>>>(Wo, wop, HH, HH);
  k_pack_pairs<<<dim3((HH * FFD) / 256), 256, 0, stream>>>(W1, w1p, HH, FFD);
  k_pack_pairs<<<dim3((FFD * HH) / 256), 256, 0, stream>>>(W2, w2p, FFD, HH);

  // 1) attention logits (WMMA over K=256, streams 1.07 GB of T with NT loads)
  k_scores<<<dim3(NN / 16, NN, BB), 32, 0, stream>>>(T, wqkp, masks, attn);
  // 2) softmax rows (B*heads*N = 16384 rows, 8 waves/block)
  k_softmax<<<dim3((BB * NHEADS * NN) / 8), 256, 0, stream>>>(attn);
  // 3) value projection (pair-packed output)
  k_vproj<<<dim3(TOK / 16, HH / 16), 32, 0, stream>>>(S, wvp, bv, vws);
  // 4) ctx = attn @ v
  k_ctx<<<dim3(NN / 16, SUBH / 16, BB * NHEADS), 32, 0, stream>>>(attn, vws, ctxws);
  // 5) attn output proj + relu + residual
  k_attnout<<<dim3(TOK / 16, HH / 16), 32, 0, stream>>>(ctxws, wop, bo, S, pre0);
  // 6) LN0
  k_layernorm<<<dim3(TOK / 8), 256, 0, stream>>>(pre0, g0, be0, xws);
  // 7) FFN up
  k_ffn1<<<dim3(TOK / 16, FFD / 16), 32, 0, stream>>>(xws, w1p, b1, h1ws);
  // 8) FFN down + residual
  k_ffn2<<<dim3(TOK / 16, HH / 16), 32, 0, stream>>>(h1ws, w2p, b2, xws, pre1);
  // 9) LN1 -> out
  k_layernorm<<<dim3(TOK / 8), 256, 0, stream>>>(pre1, g1, be1, out);
}